// Net_46059229283172
// MI455X (gfx1250) — compile-verified
//
#include <hip/hip_runtime.h>

typedef float v2f __attribute__((ext_vector_type(2)));
typedef float v8f __attribute__((ext_vector_type(8)));
typedef int   v4i __attribute__((ext_vector_type(4)));

#define Bn 512
#define Dn 256
#define Cn 10

// Async copy helper: GLOBAL_LOAD_ASYNC_TO_LDS_B128.
// Builtin signature: (AS1 v4i*, AS3 v4i*, imm offset, imm cpol).
// Integer detour for the address-space casts; low 32 bits of a flat LDS
// address are the LDS offset (aperture rules).
__device__ __forceinline__ void async_copy_b128(const void* g, void* l) {
    __builtin_amdgcn_global_load_async_to_lds_b128(
        (__attribute__((address_space(1))) v4i*)(uintptr_t)g,
        (__attribute__((address_space(3))) v4i*)(uintptr_t)l,
        0, 0);
}

// ---------------------------------------------------------------------------
// K0: init v with the analytic constant-region contribution, zero cs and hfc.
// Skipped tiles (tj >= ti+2) have y == relu(b2) everywhere.
// row i gains yb*32*max(0,6-ti); col j gains yb*32*max(0,tj-1).
// ---------------------------------------------------------------------------
__global__ void init_kernel(const float* __restrict__ b2,
                            float* __restrict__ v,
                            float* __restrict__ cs,
                            float* __restrict__ hfc) {
    int n = blockIdx.x * 256 + threadIdx.x;     // 0 .. B*D-1
    float yb = fmaxf(b2[0], 0.f);
    int col = n & (Dn - 1);
    int tt  = col >> 5;                          // tile index of this row/col
    float rowcnt = (float)max(0, 6 - tt);
    float colcnt = (float)max(0, tt - 1);
    v[n]  = yb * 32.f * (rowcnt + colcnt);
    cs[n] = 0.f;
    if (n < 4) hfc[n] = 0.f;
}

// ---------------------------------------------------------------------------
// K1: fused rs/cs. One block per (batch, 32-row strip).
// The strip (32x256 f32 = 32 KB) is moved HBM -> LDS with
// GLOBAL_LOAD_ASYNC_TO_LDS_B128 (ASYNCcnt-tracked, no VGPR round-trip),
// then reduced both ways out of LDS:
//   - column sums: tile[r*264 + t], lanes consecutive -> conflict-free
//   - row sums:    bank(r,seg) = 8r+seg -> 32 distinct banks, conflict-free
// ---------------------------------------------------------------------------
__global__ void rscs_kernel(const float* __restrict__ x,
                            float* __restrict__ rs,
                            float* __restrict__ cs) {
    int b = blockIdx.x, s = blockIdx.y;
    const float* xb = x + (size_t)b * (Dn * Dn) + (size_t)s * 32 * Dn;
    __shared__ __align__(16) float tile[32 * 264];
    int t = threadIdx.x;

    // 8 iters x 256 threads x 16B = 32 KB strip
    int rowb = t >> 6;            // 0..3
    int colb = (t & 63) * 4;      // 0,4,...,252
#pragma unroll
    for (int i = 0; i < 8; ++i) {
        int row = i * 4 + rowb;
        async_copy_b128(xb + row * Dn + colb, &tile[row * 264 + colb]);
    }
    __builtin_amdgcn_s_wait_asynccnt(0);
    __syncthreads();

    // column sums (over rows) -> cs[b, t]
    float cacc = 0.f;
#pragma unroll 8
    for (int r = 0; r < 32; ++r) cacc += tile[r * 264 + t];
    atomicAdd(&cs[b * Dn + t], cacc);

    // row sums -> rs[b, s*32 + rl]
    int rl = t >> 3, seg = t & 7;
    float p = 0.f;
#pragma unroll
    for (int q = 0; q < 32; ++q) p += tile[rl * 264 + seg + 8 * q];
    p += __shfl_xor(p, 1);
    p += __shfl_xor(p, 2);
    p += __shfl_xor(p, 4);
    if (seg == 0) rs[b * Dn + s * 32 + rl] = p;
}

// ---------------------------------------------------------------------------
// K2: fused masked conv + row/col sums. One block per (batch, 32x32 tile with
// tj <= ti+1). Stage A: WMMA f32 16x16x4 computes T[tap][pixel] =
// sum_c w2[c,tap] * relu(w1[c]*s + b1[c]) over the 34x34 halo (mask folded in).
// Stage B: y = relu(b2 + 9 shifted taps), LDS-atomic row/col partials,
// global float atomics into v.
// ---------------------------------------------------------------------------
__global__ void conv_kernel(const float* __restrict__ rs,
                            const float* __restrict__ cs,
                            const float* __restrict__ w1,
                            const float* __restrict__ b1,
                            const float* __restrict__ w2,
                            const float* __restrict__ b2,
                            float* __restrict__ v) {
    // 43 processed tiles per batch; prefix offsets per tile-row ti
    const int offs[9] = {0, 2, 5, 9, 14, 20, 27, 35, 43};
    int bid = blockIdx.x;
    int b = bid / 43, t43 = bid % 43;
    int ti = 0;
    while (t43 >= offs[ti + 1]) ++ti;
    int tj = t43 - offs[ti];
    int i0 = ti * 32, j0 = tj * 32;

    __shared__ __align__(16) float T[1168 * 12];   // [pixel][tap(0..8)+pad]
    __shared__ float rsh[36], csh[34];
    __shared__ float w1s[16], b1s[16];
    __shared__ float rowacc[32], colacc[32];

    int t = threadIdx.x;
    if (t < 36) {
        int gi = i0 - 1 + t;
        rsh[t] = (gi >= 0 && gi < Dn) ? rs[b * Dn + gi] : 0.f;
    }
    if (t >= 64 && t < 98) {
        int gj = j0 - 1 + (t - 64);
        csh[t - 64] = (gj >= 0 && gj < Dn) ? cs[b * Dn + gj] : 0.f;
    }
    if (t >= 128 && t < 144) {
        int c = t - 128;
        w1s[c] = (c < Cn) ? w1[c] : 0.f;
        b1s[c] = (c < Cn) ? b1[c] : 0.f;
    }
    if (t >= 160 && t < 192) rowacc[t - 160] = 0.f;
    if (t >= 192 && t < 224) colacc[t - 192] = 0.f;
    float bb2 = b2[0];
    __syncthreads();

    int wave  = t >> 5;
    int lane  = t & 31;
    int pix16 = lane & 15;         // M/N index within WMMA
    int hi    = lane >> 4;         // lane-half selects K sub-pair

    // A matrix (constant): A[tap][k] = w2[c*9 + tap], c = k + 4*chunk
    v2f A0, A1, A2;
    {
        int tap = pix16;
#pragma unroll
        for (int e = 0; e < 2; ++e) {
            int c0 = e + 2 * hi;
            A0[e] = (tap < 9 && (c0 + 0) < Cn) ? w2[(c0 + 0) * 9 + tap] : 0.f;
            A1[e] = (tap < 9 && (c0 + 4) < Cn) ? w2[(c0 + 4) * 9 + tap] : 0.f;
            A2[e] = (tap < 9 && (c0 + 8) < Cn) ? w2[(c0 + 8) * 9 + tap] : 0.f;
        }
    }
    // per-lane channel affine params (zero-padded beyond C)
    float w1a[6], b1a[6];
#pragma unroll
    for (int q = 0; q < 3; ++q)
#pragma unroll
        for (int e = 0; e < 2; ++e) {
            int c = e + 2 * hi + 4 * q;
            w1a[2 * q + e] = w1s[c];
            b1a[2 * q + e] = b1s[c];
        }

    // ---- Stage A: 73 groups of 16 halo pixels; 3 WMMAs each ----
    for (int g = wave; g < 73; g += 8) {
        int p   = g * 16 + pix16;
        int row = p / 34;
        int col = p - row * 34;
        int gi = i0 - 1 + row;
        int gj = j0 - 1 + col;
        bool ok = (p < 1156) && (gi >= 0) && (gi < Dn) &&
                  (gj >= 0) && (gj < Dn) && (gi >= gj);   // tril mask + padding
        float s = rsh[row] + csh[col];
        v2f B0, B1, B2;
        B0[0] = ok ? fmaxf(fmaf(w1a[0], s, b1a[0]), 0.f) : 0.f;
        B0[1] = ok ? fmaxf(fmaf(w1a[1], s, b1a[1]), 0.f) : 0.f;
        B1[0] = ok ? fmaxf(fmaf(w1a[2], s, b1a[2]), 0.f) : 0.f;
        B1[1] = ok ? fmaxf(fmaf(w1a[3], s, b1a[3]), 0.f) : 0.f;
        B2[0] = ok ? fmaxf(fmaf(w1a[4], s, b1a[4]), 0.f) : 0.f;
        B2[1] = ok ? fmaxf(fmaf(w1a[5], s, b1a[5]), 0.f) : 0.f;
        v8f acc = {0.f, 0.f, 0.f, 0.f, 0.f, 0.f, 0.f, 0.f};
        acc = __builtin_amdgcn_wmma_f32_16x16x4_f32(false, A0, false, B0, (short)0, acc, false, false);
        acc = __builtin_amdgcn_wmma_f32_16x16x4_f32(false, A1, false, B1, (short)0, acc, false, false);
        acc = __builtin_amdgcn_wmma_f32_16x16x4_f32(false, A2, false, B2, (short)0, acc, false, false);
        // D layout: lanes 0-15 hold taps 0-7 (vgpr=tap), lanes 16-31 hold taps 8-15
        float* tp = &T[p * 12];
        if (hi == 0) {
            float4 lo = make_float4(acc[0], acc[1], acc[2], acc[3]);
            float4 hi4 = make_float4(acc[4], acc[5], acc[6], acc[7]);
            *(float4*)(tp)     = lo;
            *(float4*)(tp + 4) = hi4;
        } else {
            tp[8] = acc[0];     // tap 8
        }
    }
    __syncthreads();

    // ---- Stage B: 32x32 outputs, 4 per thread ----
#pragma unroll
    for (int u = 0; u < 4; ++u) {
        int q = t + 256 * u;
        int r = q >> 5, c = q & 31;
        float y = bb2;
#pragma unroll
        for (int k = 0; k < 9; ++k) {
            int hp = (r + k / 3) * 34 + (c + k % 3);
            y += T[hp * 12 + k];
        }
        y = fmaxf(y, 0.f);
        atomicAdd(&rowacc[r], y);
        atomicAdd(&colacc[c], y);
    }
    __syncthreads();
    if (t < 32)
        atomicAdd(&v[b * Dn + i0 + t], rowacc[t]);
    else if (t < 64)
        atomicAdd(&v[b * Dn + j0 + (t - 32)], colacc[t - 32]);
}

// ---------------------------------------------------------------------------
// K3: fc1_w [4 x 131072] @ v -> 4 partial dots (atomic accumulate)
// ---------------------------------------------------------------------------
__global__ void fc1_kernel(const float* __restrict__ fc1_w,
                           const float* __restrict__ v,
                           float* __restrict__ hfc) {
    const int N = Bn * Dn;
    int idx0 = blockIdx.x * 256 + threadIdx.x;
    float a0 = 0.f, a1 = 0.f, a2 = 0.f, a3 = 0.f;
    for (int idx = idx0; idx < N; idx += 128 * 256) {
        float vv = v[idx];
        a0 = fmaf(fc1_w[idx],         vv, a0);
        a1 = fmaf(fc1_w[N + idx],     vv, a1);
        a2 = fmaf(fc1_w[2 * N + idx], vv, a2);
        a3 = fmaf(fc1_w[3 * N + idx], vv, a3);
    }
#pragma unroll
    for (int m = 16; m >= 1; m >>= 1) {
        a0 += __shfl_xor(a0, m);
        a1 += __shfl_xor(a1, m);
        a2 += __shfl_xor(a2, m);
        a3 += __shfl_xor(a3, m);
    }
    if ((threadIdx.x & 31) == 0) {
        atomicAdd(&hfc[0], a0);
        atomicAdd(&hfc[1], a1);
        atomicAdd(&hfc[2], a2);
        atomicAdd(&hfc[3], a3);
    }
}

// ---------------------------------------------------------------------------
// K4: relu + fc2 (2 outputs)
// ---------------------------------------------------------------------------
__global__ void fc2_kernel(const float* __restrict__ hfc,
                           const float* __restrict__ fc1_b,
                           const float* __restrict__ fc2_w,
                           const float* __restrict__ fc2_b,
                           float* __restrict__ out) {
    if (threadIdx.x == 0) {
        float h[4];
#pragma unroll
        for (int k = 0; k < 4; ++k) h[k] = fmaxf(hfc[k] + fc1_b[k], 0.f);
#pragma unroll
        for (int j = 0; j < 2; ++j) {
            float o = fc2_b[j];
#pragma unroll
            for (int k = 0; k < 4; ++k) o = fmaf(fc2_w[j * 4 + k], h[k], o);
            out[j] = o;
        }
    }
}

extern "C" void kernel_launch(void* const* d_in, const int* in_sizes, int n_in,
                              void* d_out, int out_size, void* d_ws, size_t ws_size,
                              hipStream_t stream) {
    const float* x     = (const float*)d_in[0];
    const float* w1    = (const float*)d_in[1];
    const float* b1    = (const float*)d_in[2];
    const float* w2    = (const float*)d_in[3];
    const float* b2    = (const float*)d_in[4];
    const float* fc1_w = (const float*)d_in[5];
    const float* fc1_b = (const float*)d_in[6];
    const float* fc2_w = (const float*)d_in[7];
    const float* fc2_b = (const float*)d_in[8];
    float* out = (float*)d_out;

    float* rs  = (float*)d_ws;           // [B*D]
    float* cs  = rs + Bn * Dn;           // [B*D]
    float* v   = cs + Bn * Dn;           // [B*D]
    float* hfc = v + Bn * Dn;            // [4]

    init_kernel<<<(Bn * Dn) / 256, 256, 0, stream>>>(b2, v, cs, hfc);
    rscs_kernel<<<dim3(Bn, Dn / 32), 256, 0, stream>>>(x, rs, cs);
    conv_kernel<<<Bn * 43, 256, 0, stream>>>(rs, cs, w1, b1, w2, b2, v);
    fc1_kernel<<<128, 256, 0, stream>>>(fc1_w, v, hfc);
    fc2_kernel<<<1, 32, 0, stream>>>(hfc, fc1_b, fc2_w, fc2_b, out);
}